// CenterWoParamMultiCosineLoss_15917148799629
// MI455X (gfx1250) — compile-verified
//
#include <hip/hip_runtime.h>

typedef __attribute__((ext_vector_type(2))) float v2f;
typedef __attribute__((ext_vector_type(8))) float v8f;

#define EPSF 1e-12f
#define B_N 8192
#define D_N 1024
#define C_N 90
#define K_N 16

// ---------------------------------------------------------------------------
// Kernel 1: per-row inverse L2 norm of centers: invc[c*K+k] = 1/sqrt(sum c^2 + eps)
// One block per (c,k) row of D=1024 floats; 256 threads load float4 each.
// ---------------------------------------------------------------------------
__global__ void center_invnorm_kernel(const float* __restrict__ centers,
                                      float* __restrict__ invc) {
    __shared__ float red[256];
    const int row = blockIdx.x;        // 0 .. C*K-1
    const int t   = threadIdx.x;       // 0 .. 255
    const float4* p = (const float4*)(centers + (size_t)row * D_N);
    float4 v = p[t];
    float ss = v.x * v.x + v.y * v.y + v.z * v.z + v.w * v.w;
    red[t] = ss;
    __syncthreads();
    for (int s = 128; s > 0; s >>= 1) {
        if (t < s) red[t] += red[t + s];
        __syncthreads();
    }
    if (t == 0) invc[row] = rsqrtf(red[0] + EPSF);
}

// ---------------------------------------------------------------------------
// Kernel 2: one wave32 per sample. Stage x in LDS + sum-of-squares, then
// 256 x V_WMMA_F32_16X16X4_F32 accumulating dot(center_k, x) for k=0..15,
// with x broadcast into all 16 B-columns. Finish with S - Q/S per sample.
// ---------------------------------------------------------------------------
__global__ void per_sample_kernel(const float* __restrict__ x,
                                  const int* __restrict__ labels,
                                  const float* __restrict__ centers,
                                  const float* __restrict__ invc,
                                  float* __restrict__ loss) {
    __shared__ float xs[8 * D_N];                 // 8 waves * 4KB = 32KB
    const int wave = threadIdx.x >> 5;
    const int lane = threadIdx.x & 31;
    const int b    = blockIdx.x * 8 + wave;       // sample index

    // ---- pass 1: load x -> LDS, accumulate sum of squares -----------------
    float ss = 0.f;
    const float4* xp = (const float4*)(x + (size_t)b * D_N);
    float4* xl = (float4*)(xs + wave * D_N);
#pragma unroll
    for (int c = 0; c < 8; ++c) {
        float4 v = xp[c * 32 + lane];
        xl[c * 32 + lane] = v;
        ss += v.x * v.x + v.y * v.y + v.z * v.z + v.w * v.w;
    }
#pragma unroll
    for (int m = 16; m >= 1; m >>= 1) ss += __shfl_xor(ss, m, 32);
    const float invx = rsqrtf(ss + EPSF);
    __syncthreads();

    // ---- pass 2: WMMA f32 16x16x4 over D in chunks of 4 -------------------
    const int label = labels[b];
    const float* cbase = centers + (size_t)label * (K_N * D_N);
    const int row   = lane & 15;                  // A-matrix M index
    const int khalf = lane >> 4;                  // 0: K=0..1, 1: K=2..3
    const float* arow = cbase + row * D_N + 2 * khalf;
    const float* xrow = xs + wave * D_N + 2 * khalf;

    v8f acc = {};
#pragma unroll 4
    for (int d0 = 0; d0 < D_N; d0 += 4) {
        v2f a  = *(const v2f*)(arow + d0);        // global_load_b64 (centers)
        v2f bb = *(const v2f*)(xrow + d0);        // ds_load_b64 (x broadcast cols)
        acc = __builtin_amdgcn_wmma_f32_16x16x4_f32(
            /*neg_a=*/false, a, /*neg_b=*/false, bb,
            /*c_mod=*/(short)0, acc, /*reuse_a=*/false, /*reuse_b=*/false);
    }

    // ---- combine: lanes 0-15 hold k=0..7 (VGPR j -> M=j), lanes 16-31 k=8..15
    const float* icp = invc + label * K_N + khalf * 8;
    float S = 0.f, Q = 0.f;
#pragma unroll
    for (int j = 0; j < 8; ++j) {
        float dk = 1.0f - acc[j] * invx * icp[j]; // d_k = 1 - x_hat . c_hat_k
        S += dk;
        Q += dk * dk;
    }
    S += __shfl_xor(S, 16, 32);
    Q += __shfl_xor(Q, 16, 32);
    // sum_k (1 - d_k/S) * d_k = S - Q/S
    if (lane == 0) loss[b] = S - Q / S;
}

// ---------------------------------------------------------------------------
// Kernel 3: deterministic tree reduction of 8192 per-sample losses -> mean.
// ---------------------------------------------------------------------------
__global__ void reduce_mean_kernel(const float* __restrict__ loss,
                                   float* __restrict__ out) {
    __shared__ float red[256];
    const int t = threadIdx.x;
    float s = 0.f;
    for (int i = t; i < B_N; i += 256) s += loss[i];
    red[t] = s;
    __syncthreads();
    for (int k = 128; k > 0; k >>= 1) {
        if (t < k) red[t] += red[t + k];
        __syncthreads();
    }
    if (t == 0) out[0] = red[0] / (float)B_N;
}

// ---------------------------------------------------------------------------
extern "C" void kernel_launch(void* const* d_in, const int* in_sizes, int n_in,
                              void* d_out, int out_size, void* d_ws, size_t ws_size,
                              hipStream_t stream) {
    const float* x       = (const float*)d_in[0];
    const int*   labels  = (const int*)d_in[1];
    const float* centers = (const float*)d_in[2];

    float* w    = (float*)d_ws;
    float* invc = w;          // C_N*K_N = 1440 floats
    float* loss = w + 4096;   // B_N = 8192 floats

    center_invnorm_kernel<<<C_N * K_N, 256, 0, stream>>>(centers, invc);
    per_sample_kernel<<<B_N / 8, 256, 0, stream>>>(x, labels, centers, invc, loss);
    reduce_mean_kernel<<<1, 256, 0, stream>>>(loss, (float*)d_out);
}